// MultiOmicsEmbedding_17171279250040
// MI455X (gfx1250) — compile-verified
//
#include <hip/hip_runtime.h>
#include <cstdint>
#include <cstddef>

typedef __attribute__((ext_vector_type(2))) float v2f;
typedef __attribute__((ext_vector_type(8))) float v8f;
typedef __attribute__((ext_vector_type(4))) int   v4i;

typedef __attribute__((address_space(1))) v4i g_v4i;  // global
typedef __attribute__((address_space(3))) v4i l_v4i;  // LDS

__device__ __forceinline__ v8f wmma_f32_4(v2f a, v2f b, v8f c) {
    // D = A(16x4) * B(4x16) + C(16x16), fp32, wave32
    return __builtin_amdgcn_wmma_f32_16x16x4_f32(false, a, false, b, (short)0, c,
                                                 false, false);
}

__device__ __forceinline__ float silu_f(float x) { return x / (1.0f + __expf(-x)); }

// async gather of one 16B chunk: global (AS1) -> LDS (AS3), tracked by ASYNCcnt
__device__ __forceinline__ void async_copy16(const float* gp, float* lp) {
    __builtin_amdgcn_global_load_async_to_lds_b128((g_v4i*)gp, (l_v4i*)lp, 0, 0);
}

__device__ __forceinline__ void wait_async0() {
    asm volatile("s_wait_asynccnt 0x0" ::: "memory");
}

// ---------------------------------------------------------------------------
// Edge message-passing kernel (one modality).
//   m = [x[dst] | x[src] | ea]  (16 edges x 144)
//   h = silu(m @ W1 + b1); h = silu(h @ W2 + b2); h = silu(h)
//   atomically accumulate h into agg[dst]  (agg is N x 128)
// Workgroup = 256 threads = 8 waves. Wave w owns output columns [16w,16w+16).
// W1/W2 column-slice fragments live in registers for the whole WG lifetime.
// m_tile gather is double-buffered via GLOBAL_LOAD_ASYNC_TO_LDS_B128 so the
// next tile's scattered HBM reads overlap the current tile's WMMA work.
// ---------------------------------------------------------------------------
#define MPAD 148   // 144 padded; 148 % 64 = 20 banks -> conflict-free, 592B rows (16B aligned)
#define HPAD 132   // 128 padded; 132 % 64 = 4 banks  -> conflict-free, 528B rows (16B aligned)

__global__ __launch_bounds__(256) void edge_mp_kernel(
    const float* __restrict__ x,     // [N,64]
    const int*   __restrict__ ei,    // [2,E]  (row0 = src, row1 = dst)
    const float* __restrict__ ea,    // [E,16]
    const float* __restrict__ W1,    // [144,128]
    const float* __restrict__ b1,    // [128]
    const float* __restrict__ W2,    // [128,128]
    const float* __restrict__ b2,    // [128]
    float*       __restrict__ agg,   // [N,128] pre-zeroed
    int nEdges)
{
    __shared__ __align__(16) float m_tile[2][16 * MPAD];
    __shared__ __align__(16) float h_tile[16 * HPAD];
    __shared__ int dst_tile[2][16];

    const int tid  = threadIdx.x;
    const int lane = tid & 31;
    const int wave = tid >> 5;
    const int ln   = lane & 15;          // row (A) / column (B,D) within halves
    const int koff = (lane >> 4) << 1;   // 0 for lanes 0-15, 2 for lanes 16-31
    const int moff = (lane >> 4) << 3;   // D-matrix row offset 0 / 8
    const int ncol = (wave << 4) + ln;   // this lane's output column (0..127)

    // --- cache B-matrix fragments (weight column slice) in registers ---
    v2f w1f[36];
#pragma unroll
    for (int s = 0; s < 36; ++s) {
        const int k = 4 * s + koff;
        w1f[s] = (v2f){ W1[k * 128 + ncol], W1[(k + 1) * 128 + ncol] };
    }
    v2f w2f[32];
#pragma unroll
    for (int s = 0; s < 32; ++s) {
        const int k = 4 * s + koff;
        w2f[s] = (v2f){ W2[k * 128 + ncol], W2[(k + 1) * 128 + ncol] };
    }
    const float bias1 = b1[ncol];
    const float bias2 = b2[ncol];

    const int nTiles = nEdges >> 4;
    const int stride = gridDim.x;

    // issue async gather of tile t (16 edges x 36 float4 chunks) into buffer b
    auto issue_gather = [&](int t, int b) {
        const int e0 = t << 4;
        if (tid < 16) dst_tile[b][tid] = ei[nEdges + e0 + tid];
        for (int idx = tid; idx < 16 * 36; idx += 256) {
            const int e = idx / 36;
            const int c = idx - e * 36;
            const int g = e0 + e;
            const float* gp;
            if (c < 16) {
                gp = x + (size_t)ei[nEdges + g] * 64 + c * 4;
            } else if (c < 32) {
                gp = x + (size_t)ei[g] * 64 + (c - 16) * 4;
            } else {
                gp = ea + (size_t)g * 16 + (c - 32) * 4;
            }
            async_copy16(gp, &m_tile[b][e * MPAD + c * 4]);
        }
    };

    int t = blockIdx.x;
    int buf = 0;
    if (t < nTiles) issue_gather(t, 0);

    for (; t < nTiles; t += stride) {
        wait_async0();        // this wave's async writes to LDS landed
        __syncthreads();      // all waves' writes landed; also protects h_tile WAR

        const int tn = t + stride;
        if (tn < nTiles) issue_gather(tn, buf ^ 1);   // overlap next gather

        // layer 1: [16x144] @ [144x(16-col slice)]
        v8f acc = (v8f){0.f, 0.f, 0.f, 0.f, 0.f, 0.f, 0.f, 0.f};
#pragma unroll
        for (int s = 0; s < 36; ++s) {
            v2f a = *(const v2f*)&m_tile[buf][ln * MPAD + 4 * s + koff];
            acc = wmma_f32_4(a, w1f[s], acc);
        }
        // silu + stage h slice for layer-2 A fragments
#pragma unroll
        for (int r = 0; r < 8; ++r) {
            const int row = r + moff;
            h_tile[row * HPAD + ncol] = silu_f(acc[r] + bias1);
        }
        __syncthreads();

        // layer 2: [16x128] @ [128x(16-col slice)]
        v8f acc2 = (v8f){0.f, 0.f, 0.f, 0.f, 0.f, 0.f, 0.f, 0.f};
#pragma unroll
        for (int s = 0; s < 32; ++s) {
            v2f a = *(const v2f*)&h_tile[ln * HPAD + 4 * s + koff];
            acc2 = wmma_f32_4(a, w2f[s], acc2);
        }
        // double silu + scatter-add into agg[dst]
#pragma unroll
        for (int r = 0; r < 8; ++r) {
            const int row = r + moff;
            const float hv = silu_f(silu_f(acc2[r] + bias2));
            const int d = dst_tile[buf][row];
            unsafeAtomicAdd(&agg[(size_t)d * 128 + ncol], hv);
        }
        buf ^= 1;
    }
}

// ---------------------------------------------------------------------------
// Generic row-major GEMM: out = act(A[nrows,K] @ W[K,ncols] + bias)
// grid.y * 128 must equal ncols. Wave w of block handles columns
// [(blockIdx.y*8 + w)*16, +16). Weights cached in registers.
// ---------------------------------------------------------------------------
template <int K, bool SILU>
__global__ __launch_bounds__(256) void row_gemm_kernel(
    const float* __restrict__ A,
    const float* __restrict__ W,
    const float* __restrict__ bias,
    float*       __restrict__ out,
    int nrows, int ncols)
{
    constexpr int KSTEPS = K / 4;
    constexpr int KPAD   = K + 4;     // K+4 keeps rows 16B aligned, banks conflict-free
    __shared__ __align__(16) float a_tile[16 * KPAD];

    const int tid  = threadIdx.x;
    const int lane = tid & 31;
    const int wave = tid >> 5;
    const int ln   = lane & 15;
    const int koff = (lane >> 4) << 1;
    const int moff = (lane >> 4) << 3;
    const int ncolg = (blockIdx.y * 8 + wave) * 16 + ln;

    v2f wf[KSTEPS];
#pragma unroll
    for (int s = 0; s < KSTEPS; ++s) {
        const int k = 4 * s + koff;
        wf[s] = (v2f){ W[(size_t)k * ncols + ncolg], W[(size_t)(k + 1) * ncols + ncolg] };
    }
    const float bv = bias[ncolg];

    const int nTiles = nrows >> 4;
    for (int rt = blockIdx.x; rt < nTiles; rt += gridDim.x) {
        // coalesced stage of 16 rows x K floats
        for (int idx = tid; idx < 16 * (K / 4); idx += 256) {
            const int e = idx / (K / 4);
            const int c = idx - e * (K / 4);
            float4 v = ((const float4*)A)[(size_t)(rt * 16 + e) * (K / 4) + c];
            *(float4*)&a_tile[e * KPAD + c * 4] = v;
        }
        __syncthreads();

        v8f acc = (v8f){0.f, 0.f, 0.f, 0.f, 0.f, 0.f, 0.f, 0.f};
#pragma unroll
        for (int s = 0; s < KSTEPS; ++s) {
            v2f a = *(const v2f*)&a_tile[ln * KPAD + 4 * s + koff];
            acc = wmma_f32_4(a, wf[s], acc);
        }
#pragma unroll
        for (int r = 0; r < 8; ++r) {
            const int row = rt * 16 + r + moff;
            float v = acc[r] + bv;
            if (SILU) v = silu_f(v);
            out[(size_t)row * ncols + ncolg] = v;
        }
        __syncthreads();
    }
}

__global__ void zero_kernel(float4* __restrict__ p, long n4) {
    for (long i = (long)blockIdx.x * blockDim.x + threadIdx.x; i < n4;
         i += (long)gridDim.x * blockDim.x) {
        p[i] = make_float4(0.f, 0.f, 0.f, 0.f);
    }
}

// ---------------------------------------------------------------------------
extern "C" void kernel_launch(void* const* d_in, const int* in_sizes, int n_in,
                              void* d_out, int out_size, void* d_ws, size_t ws_size,
                              hipStream_t stream) {
    const float* x0     = (const float*)d_in[0];
    const int*   ei0    = (const int*)  d_in[1];
    const float* ea0    = (const float*)d_in[2];
    const float* x1     = (const float*)d_in[3];
    const int*   ei1    = (const int*)  d_in[4];
    const float* ea1    = (const float*)d_in[5];
    const float* cell_x = (const float*)d_in[6];
    const float* W1_0 = (const float*)d_in[7];
    const float* b1_0 = (const float*)d_in[8];
    const float* W2_0 = (const float*)d_in[9];
    const float* b2_0 = (const float*)d_in[10];
    const float* Wu_0 = (const float*)d_in[11];
    const float* bu_0 = (const float*)d_in[12];
    const float* W1_1 = (const float*)d_in[13];
    const float* b1_1 = (const float*)d_in[14];
    const float* W2_1 = (const float*)d_in[15];
    const float* b2_1 = (const float*)d_in[16];
    const float* Wu_1 = (const float*)d_in[17];
    const float* bu_1 = (const float*)d_in[18];
    const float* Wc   = (const float*)d_in[19];
    const float* bc   = (const float*)d_in[20];

    const int N = in_sizes[0] / 64;   // 50000
    const int E = in_sizes[1] / 2;    // 500000

    float* out  = (float*)d_out;
    float* agg0 = (float*)d_ws;                      // N x 128
    float* agg1 = agg0 + (size_t)N * 128;            // N x 128

    // zero the aggregation buffers
    {
        const long n4 = (long)N * 64;  // 2*N*128 floats / 4
        zero_kernel<<<4096, 256, 0, stream>>>((float4*)d_ws, n4);
    }

    // edge message passing + scatter-add (both modalities)
    edge_mp_kernel<<<2048, 256, 0, stream>>>(x0, ei0, ea0, W1_0, b1_0, W2_0, b2_0,
                                             agg0, E);
    edge_mp_kernel<<<2048, 256, 0, stream>>>(x1, ei1, ea1, W1_1, b1_1, W2_1, b2_1,
                                             agg1, E);

    // H_i = agg_i @ Wu_i + bu_i   (no activation)
    row_gemm_kernel<128, false><<<dim3(N / 16, 1), 256, 0, stream>>>(
        agg0, Wu_0, bu_0, out, N, 128);
    row_gemm_kernel<128, false><<<dim3(N / 16, 1), 256, 0, stream>>>(
        agg1, Wu_1, bu_1, out + (size_t)N * 128, N, 128);

    // C = silu(cell_x @ Wc + bc)
    row_gemm_kernel<256, true><<<dim3(N / 16, 2), 256, 0, stream>>>(
        cell_x, Wc, bc, out + (size_t)2 * N * 128, N, 256);
}